// KalmanFilter_21036749816101
// MI455X (gfx1250) — compile-verified
//
#include <hip/hip_runtime.h>
#include <hip/hip_bf16.h>
#include <math.h>

// ---------------------------------------------------------------------------
// Kalman filter, B=512 T=2048 D=64 (derived from in_sizes at launch).
// Key observation: 'pred' uses x_t (input), not the carried state, so the
// time recursion only carries the covariance P, which is data-independent.
// Phase 1: one block computes L=chol(Q) and the gain sequence K_t into d_ws.
// Phase 2: fully parallel over (b,t): 5 row-matmuls per element done with
//          v_wmma_f32_16x16x4_f32 on 16x64 tiles per wave.
// ---------------------------------------------------------------------------

typedef float v2f __attribute__((ext_vector_type(2)));
typedef float v8f __attribute__((ext_vector_type(8)));

__device__ __forceinline__ v8f v8f_zero() {
  v8f z = {0.f, 0.f, 0.f, 0.f, 0.f, 0.f, 0.f, 0.f};
  return z;
}

__device__ __forceinline__ v8f wmma4(v2f a, v2f b, v8f c) {
  // V_WMMA_F32_16X16X4_F32 : D = A(16x4) x B(4x16) + C(16x16)
  return __builtin_amdgcn_wmma_f32_16x16x4_f32(
      /*neg_a=*/false, a, /*neg_b=*/false, b,
      /*c_mod=*/(short)0, c, /*reuse_a=*/false, /*reuse_b=*/false);
}

#define LP 65  // LDS row pitch (floats): 65 -> consecutive rows shift banks by 1

// acc[4] += Xt(16x64, LDS pitch LP)  @  W^T   with W(64x64) row-major in LDS.
// B-operand element (k,n) = W[n*LP + k]   (since (W^T)[k][n] = W[n][k]).
__device__ __forceinline__ void mm16x64_AxWT(const float* Xt, const float* W,
                                             v8f acc[4], int lane) {
  const int m = lane & 15;            // A: row per lane ; B: col per lane
  const int koff = (lane >> 4) << 1;  // lanes 16-31 hold K+2 (ISA 7.12.2)
#pragma unroll
  for (int kb = 0; kb < 64; kb += 4) {
    const int k = kb + koff;
    v2f a;
    a.x = Xt[m * LP + k];
    a.y = Xt[m * LP + k + 1];
#pragma unroll
    for (int nt = 0; nt < 4; ++nt) {
      const int n = nt * 16 + m;
      v2f b;
      b.x = W[n * LP + k];
      b.y = W[n * LP + k + 1];
      acc[nt] = wmma4(a, b, acc[nt]);
    }
  }
}

// Store C/D-layout accumulators as a row-major 16x64 tile (pitch LP) in LDS.
// C/D layout: lane L holds col (L&15); VGPR r holds row r + 8*(L>=16).
__device__ __forceinline__ void store_tile(float* T, const v8f acc[4], int lane) {
  const int c = lane & 15;
  const int rh = (lane >> 4) * 8;
#pragma unroll
  for (int nt = 0; nt < 4; ++nt)
#pragma unroll
    for (int r = 0; r < 8; ++r)
      T[(rh + r) * LP + nt * 16 + c] = acc[nt][r];
}

// ---------------------------------------------------------------------------
// Phase 1: L = chol(Q); K_t recursion (data-independent, one workgroup).
// ---------------------------------------------------------------------------
__device__ __forceinline__ void mm_nn64(const float* A, const float* Bm, float* C,
                                        int tid) {  // C = A @ B
  for (int idx = tid; idx < 4096; idx += 256) {
    int i = idx >> 6, j = idx & 63;
    float s = 0.f;
    for (int k = 0; k < 64; ++k) s = fmaf(A[i * 64 + k], Bm[k * 64 + j], s);
    C[idx] = s;
  }
}
__device__ __forceinline__ void mm_nt64(const float* A, const float* Bm, float* C,
                                        int tid) {  // C = A @ B^T
  for (int idx = tid; idx < 4096; idx += 256) {
    int i = idx >> 6, j = idx & 63;
    float s = 0.f;
    for (int k = 0; k < 64; ++k) s = fmaf(A[i * 64 + k], Bm[j * 64 + k], s);
    C[idx] = s;
  }
}

__global__ __launch_bounds__(256) void kf_gains(const float* __restrict__ Q,
                                                const float* __restrict__ R,
                                                float* __restrict__ gains,
                                                float* __restrict__ Lq, int Tlen) {
  extern __shared__ float sm[];
  float* sQ = sm;            // 4096
  float* sR = sQ + 4096;     // 4096
  float* sP = sR + 4096;     // 4096
  float* sA = sP + 4096;     // 4096 (scratch / chol / K)
  float* sB = sA + 4096;     // 4096 (P_pred)
  float* sC = sB + 4096;     // 4096 (scratch)
  float* AUG = sC + 4096;    // 64*128 Gauss-Jordan
  float* fcol = AUG + 8192;  // 64
  const int tid = threadIdx.x;

  for (int i = tid; i < 4096; i += 256) {
    sQ[i] = Q[i];
    sR[i] = R[i];
    sP[i] = ((i >> 6) == (i & 63)) ? 1.f : 0.f;  // P0 = I
    sA[i] = Q[i];                                // chol workspace
  }
  __syncthreads();

  // In-place right-looking Cholesky of sA (lower).
  for (int j = 0; j < 64; ++j) {
    if (tid == 0) sA[j * 64 + j] = sqrtf(sA[j * 64 + j]);
    __syncthreads();
    const float d = sA[j * 64 + j];
    for (int i = j + 1 + tid; i < 64; i += 256) sA[i * 64 + j] /= d;
    __syncthreads();
    for (int idx = tid; idx < 4096; idx += 256) {
      int i = idx >> 6, k = idx & 63;
      if (i > j && k > j && k <= i) sA[i * 64 + k] -= sA[i * 64 + j] * sA[k * 64 + j];
    }
    __syncthreads();
  }
  for (int i = tid; i < 4096; i += 256) {
    int r = i >> 6, c = i & 63;
    Lq[i] = (c <= r) ? sA[i] : 0.f;
  }
  __syncthreads();

  // Riccati-style recursion producing K_t, t = 1..T-1.
  for (int t = 1; t < Tlen; ++t) {
    mm_nn64(sQ, sP, sA, tid);  __syncthreads();  // sA = Q P
    mm_nt64(sA, sQ, sB, tid);  __syncthreads();  // sB = P_pred = Q P Q^T
    mm_nt64(sB, sR, sC, tid);  __syncthreads();  // sC = P_pred R^T
    // AUG = [ S = R (P_pred R^T) | I ]
    for (int idx = tid; idx < 4096; idx += 256) {
      int i = idx >> 6, j = idx & 63;
      float s = 0.f;
      for (int k = 0; k < 64; ++k) s = fmaf(sR[i * 64 + k], sC[k * 64 + j], s);
      AUG[i * 128 + j] = s;
      AUG[i * 128 + 64 + j] = (i == j) ? 1.f : 0.f;
    }
    __syncthreads();
    // Gauss-Jordan (matches jnp.linalg.inv's garbage-on-singular behavior).
    for (int c = 0; c < 64; ++c) {
      for (int r = tid; r < 64; r += 256) fcol[r] = AUG[r * 128 + c];
      __syncthreads();
      const float piv = fcol[c];
      for (int j = tid; j < 128; j += 256) AUG[c * 128 + j] /= piv;
      __syncthreads();
      for (int idx = tid; idx < 64 * 128; idx += 256) {
        int r = idx >> 7, j = idx & 127;
        if (r != c) AUG[idx] -= fcol[r] * AUG[c * 128 + j];
      }
      __syncthreads();
    }
    // K = (P_pred R^T) S^-1 -> sA and global gains
    for (int idx = tid; idx < 4096; idx += 256) {
      int i = idx >> 6, j = idx & 63;
      float s = 0.f;
      for (int k = 0; k < 64; ++k) s = fmaf(sC[i * 64 + k], AUG[k * 128 + 64 + j], s);
      sA[idx] = s;
      gains[(size_t)(t - 1) * 4096 + idx] = s;
    }
    __syncthreads();
    mm_nn64(sR, sB, sC, tid);  __syncthreads();  // sC = R P_pred
    // P = P_pred - K (R P_pred)
    for (int idx = tid; idx < 4096; idx += 256) {
      int i = idx >> 6, j = idx & 63;
      float s = 0.f;
      for (int k = 0; k < 64; ++k) s = fmaf(sA[i * 64 + k], sC[k * 64 + j], s);
      sP[idx] = sB[idx] - s;
    }
    __syncthreads();
  }
}

// ---------------------------------------------------------------------------
// Phase 2: fully parallel WMMA data path. 128 threads = 4 waves; each wave
// owns 16 batch rows at one timestep. grid = (ceil(B/64), T).
// ---------------------------------------------------------------------------
__global__ __launch_bounds__(128) void kf_main(
    const float* __restrict__ x, const float* __restrict__ Wt,
    const float* __restrict__ bt, const float* __restrict__ Wo,
    const float* __restrict__ bo, const float* __restrict__ eps,
    const float* __restrict__ gains, const float* __restrict__ Lq,
    float* __restrict__ out, int Bsz, int Tlen) {
  extern __shared__ float sm[];
  const int t = blockIdx.y;
  const int b0 = blockIdx.x * 64;
  const int tid = threadIdx.x;
  const int wave = tid >> 5, lane = tid & 31;

  float* sWt = sm;                  // 64*LP
  float* sWo = sWt + 64 * LP;       // 64*LP
  float* sK  = sWo + 64 * LP;       // 64*LP
  float* sL  = sK + 64 * LP;        // 64*LP
  float* sBt = sL + 64 * LP;        // 64
  float* sBo = sBt + 64;            // 64
  float* wbase = sBo + 64 + wave * (4 * 16 * LP);
  float* sX = wbase;                // 16*LP  x tile
  float* sE = sX + 16 * LP;         // 16*LP  eps tile
  float* sP = sE + 16 * LP;         // 16*LP  pred tile
  float* sT = sP + 16 * LP;         // 16*LP  innov/states tile

  // Shared weights / gain / chol factor (bank-friendly pitch-65 layout).
  for (int i = tid; i < 4096; i += 128) {
    int r = i >> 6, c = i & 63;
    sWt[r * LP + c] = Wt[i];
    sWo[r * LP + c] = Wo[i];
    if (t > 0) {
      sK[r * LP + c] = gains[(size_t)(t - 1) * 4096 + i];
      sL[r * LP + c] = Lq[i];
    }
  }
  for (int i = tid; i < 64; i += 128) { sBt[i] = bt[i]; sBo[i] = bo[i]; }

  // Per-wave 16-row input tiles (coalesced 128B per iteration).
  const int brow = b0 + wave * 16;
  for (int i = lane; i < 16 * 64; i += 32) {
    int r = i >> 6, c = i & 63;
    int rb = brow + r; if (rb >= Bsz) rb = Bsz - 1;
    sX[r * LP + c] = x[((size_t)rb * Tlen + t) * 64 + c];
    if (t > 0) sE[r * LP + c] = eps[(((size_t)(t - 1)) * Bsz + rb) * 64 + c];
  }
  __syncthreads();

  const int col = lane & 15;
  const int rh = (lane >> 4) * 8;

  // pred = x @ Wt^T + b_t (+ eps @ L^T when t>0)
  v8f pred[4];
#pragma unroll
  for (int nt = 0; nt < 4; ++nt) pred[nt] = v8f_zero();
  mm16x64_AxWT(sX, sWt, pred, lane);
  if (t > 0) mm16x64_AxWT(sE, sL, pred, lane);
#pragma unroll
  for (int nt = 0; nt < 4; ++nt) {
    const float bv = sBt[nt * 16 + col];
#pragma unroll
    for (int r = 0; r < 8; ++r) pred[nt][r] += bv;
  }
  store_tile(sP, pred, lane);

  if (t > 0) {
    // innov = x - (pred @ Wo^T + b_o)
    v8f obs[4];
#pragma unroll
    for (int nt = 0; nt < 4; ++nt) obs[nt] = v8f_zero();
    mm16x64_AxWT(sP, sWo, obs, lane);
#pragma unroll
    for (int nt = 0; nt < 4; ++nt) {
      const float bv = sBo[nt * 16 + col];
#pragma unroll
      for (int r = 0; r < 8; ++r)
        obs[nt][r] = sX[(rh + r) * LP + nt * 16 + col] - (obs[nt][r] + bv);
    }
    store_tile(sT, obs, lane);

    // states = pred + innov @ K^T   (pred still live in VGPRs)
    v8f st[4];
#pragma unroll
    for (int nt = 0; nt < 4; ++nt) st[nt] = v8f_zero();
    mm16x64_AxWT(sT, sK, st, lane);
#pragma unroll
    for (int nt = 0; nt < 4; ++nt)
#pragma unroll
      for (int r = 0; r < 8; ++r) st[nt][r] += pred[nt][r];
    store_tile(sT, st, lane);
  }

  // out = states @ Wo^T + b_o
  const float* Sstate = (t > 0) ? sT : sP;
  v8f o[4];
#pragma unroll
  for (int nt = 0; nt < 4; ++nt) o[nt] = v8f_zero();
  mm16x64_AxWT(Sstate, sWo, o, lane);
#pragma unroll
  for (int nt = 0; nt < 4; ++nt) {
    const float bv = sBo[nt * 16 + col];
#pragma unroll
    for (int r = 0; r < 8; ++r) {
      const int rb = brow + rh + r;
      if (rb < Bsz)
        out[((size_t)rb * Tlen + t) * 64 + nt * 16 + col] = o[nt][r] + bv;
    }
  }
}

// ---------------------------------------------------------------------------
extern "C" void kernel_launch(void* const* d_in, const int* in_sizes, int n_in,
                              void* d_out, int out_size, void* d_ws, size_t ws_size,
                              hipStream_t stream) {
  const float* x  = (const float*)d_in[0];
  const float* Wt = (const float*)d_in[1];
  const float* bt = (const float*)d_in[2];
  const float* Q  = (const float*)d_in[3];
  const float* Wo = (const float*)d_in[4];
  const float* bo = (const float*)d_in[5];
  const float* R  = (const float*)d_in[6];
  const float* ep = (const float*)d_in[7];

  const long xs = in_sizes[0];          // B*T*64
  const long es = in_sizes[7];          // (T-1)*B*64
  const int Bsz  = (int)((xs - es) / 64);
  const int Tlen = (int)(xs / ((long)Bsz * 64));

  float* gains = (float*)d_ws;                       // (T-1)*64*64
  float* Lq = gains + (size_t)(Tlen - 1) * 4096;     // 64*64

  const size_t sh1 = (size_t)(6 * 4096 + 8192 + 64) * sizeof(float);
  kf_gains<<<1, 256, sh1, stream>>>(Q, R, gains, Lq, Tlen);

  const size_t sh2 = (size_t)(4 * 64 * LP + 128 + 4 * 4 * 16 * LP) * sizeof(float);
  dim3 grid((Bsz + 63) / 64, Tlen);
  kf_main<<<grid, 128, sh2, stream>>>(x, Wt, bt, Wo, bo, ep, gains, Lq,
                                      (float*)d_out, Bsz, Tlen);
}